// VerticalSAMS_26319559590474
// MI455X (gfx1250) — compile-verified
//
#include <hip/hip_runtime.h>

#define B_     4096
#define NFIELD 20
#define NEMB   16
#define HID    512
#define K_     16
#define GIN    320     // NFIELD * NEMB
#define EPSF   1e-10f

typedef __attribute__((ext_vector_type(2))) float v2f;
typedef __attribute__((ext_vector_type(8))) float v8f;

__device__ __forceinline__ v8f wmma_f32_16x16x4(v2f a, v2f b, v8f c) {
  // 8-arg pattern: (neg_a, A, neg_b, B, c_mod, C, reuse_a, reuse_b)
  return __builtin_amdgcn_wmma_f32_16x16x4_f32(false, a, false, b, (short)0, c,
                                               false, false);
}

// ---------------------------------------------------------------- init
__global__ void __launch_bounds__(256)
vsams_init_kernel(float* __restrict__ out, int* __restrict__ counts) {
  int t = blockIdx.x * blockDim.x + threadIdx.x;
  if (t <= B_) out[t] = 0.0f;          // y[0..B-1] and loss slot
  if (t < K_) counts[t] = 0;
}

// ---------------------------------------------------------- gate layer 1
// h = relu(sql_emb @ gw1 + gb1)   [B,320]@[320,512]
// grid: (8 n-groups of 64 cols, 256 m-tiles of 16 rows), block = 128 (4 waves)
__global__ void __launch_bounds__(128)
vsams_gate_l1_kernel(const int* __restrict__ sql,
                     const float* __restrict__ sql_emb_w,
                     const float* __restrict__ gw1,
                     const float* __restrict__ gb1,
                     float* __restrict__ h) {
  __shared__ float Alds[16 * GIN];               // 20 KB: 16 rows of sql_emb
  const int tid = threadIdx.x;
  const int m0  = blockIdx.y * 16;
  // gather sql embeddings for this row tile
  for (int idx = tid; idx < 16 * GIN; idx += 128) {
    int r = idx / GIN, c = idx - r * GIN;
    int f = c >> 4, e = c & 15;
    int s = sql[(m0 + r) * NFIELD + f];
    Alds[idx] = sql_emb_w[s * NEMB + e];
  }
  __syncthreads();
  const int lane = tid & 31, wave = tid >> 5;
  const int n0 = blockIdx.x * 64 + wave * 16;
  const int m  = lane & 15;     // A row / B,C col
  const int kh = lane >> 4;     // k half
  v8f acc = {0.f, 0.f, 0.f, 0.f, 0.f, 0.f, 0.f, 0.f};
#pragma unroll 4
  for (int k0 = 0; k0 < GIN; k0 += 4) {
    v2f a, b;
    const float* ap = &Alds[m * GIN + k0 + 2 * kh];
    a.x = ap[0]; a.y = ap[1];                     // ds_load_b64
    const float* bp = &gw1[(size_t)(k0 + 2 * kh) * HID + n0 + m];
    b.x = bp[0]; b.y = bp[HID];
    acc = wmma_f32_16x16x4(a, b, acc);
  }
#pragma unroll
  for (int v = 0; v < 8; ++v) {
    int row = m0 + v + 8 * kh;
    int col = n0 + m;
    float val = acc[v] + gb1[col];
    h[(size_t)row * HID + col] = val > 0.f ? val : 0.f;
  }
}

// ------------------------------------------------------------ gate head
// logits = h @ gw2 + gb2 ; softmax ; top-2 ; gates + expert lists
// grid: 256 blocks of 32 (one wave per 16 rows)
__global__ void __launch_bounds__(32)
vsams_gate_head_kernel(const float* __restrict__ h,
                       const float* __restrict__ gw2,
                       const float* __restrict__ gb2,
                       float* __restrict__ gates,
                       int* __restrict__ counts,
                       int* __restrict__ lists) {
  __shared__ float logit[16 * 16];
  const int lane = threadIdx.x;
  const int m0 = blockIdx.x * 16;
  const int m  = lane & 15;
  const int kh = lane >> 4;
  v8f acc = {0.f, 0.f, 0.f, 0.f, 0.f, 0.f, 0.f, 0.f};
#pragma unroll 4
  for (int k0 = 0; k0 < HID; k0 += 4) {
    v2f a, b;
    const float* ap = &h[(size_t)(m0 + m) * HID + k0 + 2 * kh];
    a.x = ap[0]; a.y = ap[1];
    const float* bp = &gw2[(k0 + 2 * kh) * K_ + m];
    b.x = bp[0]; b.y = bp[K_];
    acc = wmma_f32_16x16x4(a, b, acc);
  }
#pragma unroll
  for (int v = 0; v < 8; ++v)
    logit[(v + 8 * kh) * 16 + m] = acc[v] + gb2[m];
  __syncthreads();
  if (lane < 16) {
    const int b = m0 + lane;
    float l[K_];
    float mx = -3.0e38f;
    for (int j = 0; j < K_; ++j) { l[j] = logit[lane * 16 + j]; mx = fmaxf(mx, l[j]); }
    float s = 0.f;
    for (int j = 0; j < K_; ++j) { l[j] = __expf(l[j] - mx); s += l[j]; }
    float inv = 1.f / s;
    for (int j = 0; j < K_; ++j) l[j] *= inv;           // softmax scores
    // top-2, first-occurrence tie-break (matches jax.lax.top_k)
    int i1 = 0; float v1 = l[0];
    for (int j = 1; j < K_; ++j) if (l[j] > v1) { v1 = l[j]; i1 = j; }
    int i2 = -1; float v2 = -1.f;
    for (int j = 0; j < K_; ++j) if (j != i1 && l[j] > v2) { v2 = l[j]; i2 = j; }
    float denom = v1 + v2;
    float g1 = v1 / denom, g2 = v2 / denom;
    for (int j = 0; j < K_; ++j) gates[b * K_ + j] = 0.f;
    gates[b * K_ + i1] = g1;
    gates[b * K_ + i2] = g2;
    int p1 = atomicAdd(&counts[i1], 1);
    lists[i1 * B_ + p1] = b;
    int p2 = atomicAdd(&counts[i2], 1);
    lists[i2 * B_ + p2] = b;
  }
}

// ------------------------------------------------------------- experts
// Per (expert, 16-sample tile): eh = relu(x_emb @ ew1[k] + eb1[k]);
// eo = eh @ ew2[k] + eb2[k]; y[b] += gate * eo
// grid: 16 experts * 256 tiles = 4096 blocks of 128 (4 waves)
__global__ void __launch_bounds__(128)
vsams_expert_kernel(const int* __restrict__ x,
                    const float* __restrict__ inp_emb_w,
                    const float* __restrict__ ew1,
                    const float* __restrict__ eb1,
                    const float* __restrict__ ew2,
                    const float* __restrict__ eb2,
                    const float* __restrict__ gates,
                    const int* __restrict__ counts,
                    const int* __restrict__ lists,
                    float* __restrict__ y) {
  __shared__ float Alds[16 * GIN];   // 20 KB
  __shared__ float EH[16 * HID];     // 32 KB
  __shared__ float psum[16 * 8];
  __shared__ int   bids[16];
  const int k    = blockIdx.x >> 8;
  const int tile = blockIdx.x & 255;
  const int cnt  = counts[k];
  const int base = tile * 16;
  if (base >= cnt) return;           // block-uniform: EXEC all-1s for WMMA
  const int tid = threadIdx.x;
  if (tid < 16) bids[tid] = (base + tid < cnt) ? lists[k * B_ + base + tid] : -1;
  __syncthreads();
  // gather x_emb rows for this tile (zeros for padding rows)
  for (int idx = tid; idx < 16 * GIN; idx += 128) {
    int r = idx / GIN, c = idx - r * GIN;
    int b = bids[r];
    float val = 0.f;
    if (b >= 0) {
      int f = c >> 4, e = c & 15;
      int xi = x[b * NFIELD + f];
      val = inp_emb_w[(size_t)xi * NEMB + e];
    }
    Alds[idx] = val;
  }
  __syncthreads();
  const int lane = tid & 31, wave = tid >> 5;
  const int m  = lane & 15;
  const int kh = lane >> 4;
  const float* W1 = ew1 + (size_t)k * GIN * HID;
  const float* B1 = eb1 + k * HID;
  for (int nt = 0; nt < 8; ++nt) {
    const int n0 = (wave * 8 + nt) * 16;
    v8f acc = {0.f, 0.f, 0.f, 0.f, 0.f, 0.f, 0.f, 0.f};
#pragma unroll 4
    for (int k0 = 0; k0 < GIN; k0 += 4) {
      v2f a, b;
      const float* ap = &Alds[m * GIN + k0 + 2 * kh];
      a.x = ap[0]; a.y = ap[1];
      const float* bp = &W1[(size_t)(k0 + 2 * kh) * HID + n0 + m];
      b.x = bp[0]; b.y = bp[HID];
      acc = wmma_f32_16x16x4(a, b, acc);
    }
#pragma unroll
    for (int v = 0; v < 8; ++v) {
      int row = v + 8 * kh;
      int col = n0 + m;
      float val = acc[v] + B1[col];
      EH[row * HID + col] = val > 0.f ? val : 0.f;
    }
  }
  __syncthreads();
  // matvec: eo[row] = eh[row,:] . ew2[k,:,0]
  {
    int row = tid >> 3, seg = tid & 7;
    const float* w2 = ew2 + k * HID + seg * 64;
    const float* eh = EH + row * HID + seg * 64;
    float p = 0.f;
#pragma unroll 8
    for (int c = 0; c < 64; ++c) p += eh[c] * w2[c];
    psum[row * 8 + seg] = p;
  }
  __syncthreads();
  if (tid < 16) {
    int b = bids[tid];
    if (b >= 0) {
      float eo = eb2[k];
      for (int s = 0; s < 8; ++s) eo += psum[tid * 8 + s];
      float g = gates[b * K_ + k];
      atomicAdd(&y[b], g * eo);
    }
  }
}

// ---------------------------------------------------------------- loss
__global__ void __launch_bounds__(256)
vsams_loss_kernel(const float* __restrict__ gates, float* __restrict__ out) {
  __shared__ float simp[256], sld[256];
  const int t = threadIdx.x;
  const int k = t & 15, chunk = t >> 4;          // 16 chunks x 256 rows
  float pi = 0.f, pl = 0.f;
  for (int r = chunk * 256; r < chunk * 256 + 256; ++r) {
    float g = gates[r * K_ + k];
    pi += g;
    pl += (g > 0.f) ? 1.f : 0.f;
  }
  simp[t] = pi; sld[t] = pl;
  __syncthreads();
  if (t == 0) {
    float imp[K_], ld[K_];
    for (int kk = 0; kk < K_; ++kk) { imp[kk] = 0.f; ld[kk] = 0.f; }
    for (int c = 0; c < 16; ++c)
      for (int kk = 0; kk < K_; ++kk) {
        imp[kk] += simp[c * 16 + kk];
        ld[kk]  += sld[c * 16 + kk];
      }
    float mi = 0.f, ml = 0.f;
    for (int kk = 0; kk < K_; ++kk) { mi += imp[kk]; ml += ld[kk]; }
    mi *= (1.f / 16.f); ml *= (1.f / 16.f);
    float vi = 0.f, vl = 0.f;
    for (int kk = 0; kk < K_; ++kk) {
      float di = imp[kk] - mi, dl = ld[kk] - ml;
      vi += di * di; vl += dl * dl;
    }
    vi *= (1.f / 16.f); vl *= (1.f / 16.f);
    out[B_] = vi / (mi * mi + EPSF) + vl / (ml * ml + EPSF);
  }
}

// --------------------------------------------------------------- launch
extern "C" void kernel_launch(void* const* d_in, const int* in_sizes, int n_in,
                              void* d_out, int out_size, void* d_ws, size_t ws_size,
                              hipStream_t stream) {
  const int*   x         = (const int*)d_in[0];
  const int*   sql       = (const int*)d_in[1];
  const float* sql_emb_w = (const float*)d_in[2];
  const float* inp_emb_w = (const float*)d_in[3];
  const float* gw1 = (const float*)d_in[4];
  const float* gb1 = (const float*)d_in[5];
  const float* gw2 = (const float*)d_in[6];
  const float* gb2 = (const float*)d_in[7];
  const float* ew1 = (const float*)d_in[8];
  const float* eb1 = (const float*)d_in[9];
  const float* ew2 = (const float*)d_in[10];
  const float* eb2 = (const float*)d_in[11];
  float* out = (float*)d_out;                    // y[4096] then loss

  // workspace layout (~8.9 MB)
  float* h      = (float*)d_ws;                  // B*HID
  float* gates  = h + (size_t)B_ * HID;          // B*K
  int*   counts = (int*)(gates + (size_t)B_ * K_);
  int*   lists  = counts + K_;                   // K*B

  vsams_init_kernel<<<17, 256, 0, stream>>>(out, counts);
  vsams_gate_l1_kernel<<<dim3(8, 256), 128, 0, stream>>>(sql, sql_emb_w, gw1, gb1, h);
  vsams_gate_head_kernel<<<256, 32, 0, stream>>>(h, gw2, gb2, gates, counts, lists);
  vsams_expert_kernel<<<K_ * 256, 128, 0, stream>>>(x, inp_emb_w, ew1, eb1, ew2,
                                                    eb2, gates, counts, lists, out);
  vsams_loss_kernel<<<1, 256, 0, stream>>>(gates, out);
}